// SpikeDrivenMoE_25262997636017
// MI455X (gfx1250) — compile-verified
//
#include <hip/hip_runtime.h>
#include <hip/hip_bf16.h>

// ---------------------------------------------------------------------------
// SpikeDrivenMoE for MI455X (gfx1250, wave32).
//   T=4, N=2048 tokens, D=1024, E=8 experts, ef=512, top-2 routing.
//   GEMMs use V_WMMA_F32_16X16X4_F32 (fp32 preserved: LIF thresholding makes
//   bf16 casts flip spikes). Hidden spikes are bit-packed (4 MB scratch).
//   x + w_up + w_down (~65 MB) are L2-resident (192 MB) -> compute/L2-latency
//   bound; global_prefetch_b8 hides next-slab latency, NT stores keep the
//   33 MB output from evicting the working set.
// ---------------------------------------------------------------------------

typedef float v2f __attribute__((ext_vector_type(2)));
typedef float v8f __attribute__((ext_vector_type(8)));

#define T_STEPS 4
#define NTOK    2048
#define DDIM    1024
#define EFD     512
#define NEXP    8

__device__ __constant__ const float kBETA = 0.9f;
__device__ __constant__ const float kVTH  = 1.0f;

// hs bitmask: [t][e][n][f/32] -> T*E*N*16 u32 words = 4 MB
#define HS_WORDS (T_STEPS * NEXP * NTOK * 16)

__device__ __forceinline__ v8f wmma4(v2f a, v2f b, v8f c) {
  // D = A(16x4,f32) * B(4x16,f32) + C(16x16,f32)
  return __builtin_amdgcn_wmma_f32_16x16x4_f32(
      /*neg_a=*/false, a, /*neg_b=*/false, b,
      /*c_mod=*/(short)0, c, /*reuse_a=*/false, /*reuse_b=*/false);
}

// ---------------------------------------------------------------------------
// Zero the spike bitmask (atomicOr target) and lb statistics.
// ---------------------------------------------------------------------------
__global__ void k_init(unsigned* __restrict__ hs, float* __restrict__ stats) {
  int gid = blockIdx.x * blockDim.x + threadIdx.x;
  for (int i = gid; i < HS_WORDS; i += gridDim.x * blockDim.x) hs[i] = 0u;
  if (gid < 16) stats[gid] = 0.f;
}

// ---------------------------------------------------------------------------
// Routing: LIF over x -> firing rates -> cluster rates -> expert scores ->
// top-2 softmax weights (wroute[N][E]) + load-balance statistics.
// One wave32 per token; lane owns features d = f*32+lane (f = 0..31).
// cluster(d) = d%64, expert(d) = (d%64)>>3  ->  lane-local: even f -> expert
// lane>>3, odd f -> expert 4+(lane>>3).
// ---------------------------------------------------------------------------
__global__ __launch_bounds__(256)
void k_route(const float* __restrict__ x, const float* __restrict__ bias,
             float* __restrict__ wroute, float* __restrict__ stats) {
  const int lane = threadIdx.x & 31;
  const int wave = threadIdx.x >> 5;
  const int n = blockIdx.x * 8 + wave;

  float ce = 0.f, co = 0.f;  // spike counts for expert lane>>3 and 4+(lane>>3)
  for (int f = 0; f < 32; ++f) {
    const int d = f * 32 + lane;
    float v = 0.f, cnt = 0.f;
#pragma unroll
    for (int t = 0; t < T_STEPS; ++t) {
      v = kBETA * v + x[((size_t)t * NTOK + n) * DDIM + d];
      if (v >= kVTH) { cnt += 1.f; v -= kVTH; }
    }
    if (f & 1) co += cnt; else ce += cnt;
  }
  // reduce within the 4 groups of 8 lanes
  for (int off = 4; off >= 1; off >>= 1) {
    ce += __shfl_xor(ce, off, 32);
    co += __shfl_xor(co, off, 32);
  }
  // lanes 0..7 hold expert scores: es[e] = count/(T*16*8) = count/512 (+bias)
  float eA = __shfl(ce, (lane & 3) * 8, 32);
  float eB = __shfl(co, (lane & 3) * 8, 32);
  float es_mine = ((lane < 4) ? eA : eB) * (1.0f / 512.0f);
  if (lane < 8) es_mine += bias[lane];

  float esv[8];
#pragma unroll
  for (int j = 0; j < 8; ++j) esv[j] = __shfl(es_mine, j, 32);

  // top-2 (stable: lowest index wins ties, matching lax.top_k)
  float m1 = -1e30f, m2 = -1e30f; int i1 = 0, i2 = 0;
#pragma unroll
  for (int j = 0; j < 8; ++j) {
    float e = esv[j];
    if (e > m1)      { m2 = m1; i2 = i1; m1 = e; i1 = j; }
    else if (e > m2) { m2 = e; i2 = j; }
  }
  float r  = __expf(m2 - m1);
  float p1 = 1.f / (1.f + r);
  float p2 = r   / (1.f + r);

  float ssum = 0.f;
#pragma unroll
  for (int j = 0; j < 8; ++j) ssum += __expf(esv[j] - m1);

  if (lane < 8) {
    float w = (lane == i1 ? p1 : 0.f) + (lane == i2 ? p2 : 0.f);
    wroute[(size_t)n * NEXP + lane] = w;
    atomicAdd(&stats[8 + lane], __expf(esv[lane] - m1) / ssum);  // rp sums
  }
  if (lane == 0) {
    atomicAdd(&stats[i1], 1.f);  // top-k counts
    atomicAdd(&stats[i2], 1.f);
  }
}

// ---------------------------------------------------------------------------
// GEMM1 + LIF: h[t] = x[t] @ w_up[e], LIF over t, bit-pack spikes into hs.
// Grid: (N/32, ef/32, E). Block 128 = 4 waves, each wave a 16x16 tile.
// K = D = 1024 staged through LDS in 32-wide slabs; B fragment reused for
// all 4 timesteps; next slab prefetched into L2 during staging.
// ---------------------------------------------------------------------------
__global__ __launch_bounds__(128)
void k_up(const float* __restrict__ x, const float* __restrict__ wup,
          unsigned* __restrict__ hs) {
  __shared__ float Xs[T_STEPS][32][33];  // [t][m][k], padded
  __shared__ float Ws[32][33];           // [f][k] transposed

  const int tid  = threadIdx.x;
  const int lane = tid & 31;
  const int wave = tid >> 5;
  const int wm = wave & 1, wf = wave >> 1;
  const int e  = blockIdx.z;
  const int mb = blockIdx.x * 32;
  const int fb = blockIdx.y * 32;
  const int m0 = mb + wm * 16;
  const int row16 = lane & 15;
  const int koff  = (lane >> 4) * 2;

  v8f acc[T_STEPS];
#pragma unroll
  for (int t = 0; t < T_STEPS; ++t)
#pragma unroll
    for (int i = 0; i < 8; ++i) acc[t][i] = 0.f;

  for (int kb = 0; kb < DDIM; kb += 32) {
    __syncthreads();
    for (int idx = tid; idx < 4096; idx += 128) {        // 4 t * 32 m * 32 k
      int t = idx >> 10, rem = idx & 1023, m = rem >> 5, k = rem & 31;
      const float* p = &x[((size_t)t * NTOK + mb + m) * DDIM + kb + k];
      Xs[t][m][k] = *p;
      __builtin_prefetch(p + 32, 0, 1);                  // next K-slab -> L2
    }
    for (int idx = tid; idx < 1024; idx += 128) {        // 32 k * 32 f
      int k = idx >> 5, f = idx & 31;
      const float* p = &wup[((size_t)e * DDIM + kb + k) * EFD + fb + f];
      Ws[f][k] = *p;
      __builtin_prefetch(p + 32 * EFD, 0, 1);            // next K-slab -> L2
    }
    __syncthreads();
#pragma unroll
    for (int kk = 0; kk < 32; kk += 4) {
      v2f b;
      b.x = Ws[wf * 16 + row16][kk + koff];
      b.y = Ws[wf * 16 + row16][kk + koff + 1];
#pragma unroll
      for (int t = 0; t < T_STEPS; ++t) {
        v2f a;
        a.x = Xs[t][wm * 16 + row16][kk + koff];
        a.y = Xs[t][wm * 16 + row16][kk + koff + 1];
        acc[t] = wmma4(a, b, acc[t]);
      }
    }
  }

  // LIF over t, ballot-pack the 16x16 spike tile, OR into the global bitmask.
  // C/D layout: VGPR i, lane<16 -> (M=i, N=lane); lane>=16 -> (M=8+i, N=lane-16)
  const int wordcol = fb >> 5;                // which 32-bit f-word
  const unsigned sh = (unsigned)(wf * 16);    // 16-bit half within the word
  float vmem[8];
#pragma unroll
  for (int i = 0; i < 8; ++i) vmem[i] = 0.f;
  unsigned lowm = 0u, him = 0u;
#pragma unroll
  for (int t = 0; t < T_STEPS; ++t) {
#pragma unroll
    for (int i = 0; i < 8; ++i) {
      float v = kBETA * vmem[i] + acc[t][i];
      int s = (v >= kVTH);
      vmem[i] = v - (s ? kVTH : 0.f);
      unsigned m32 = (unsigned)__ballot(s);
      if (lane == i)      lowm = m32;   // lanes 0..7 keep row m0+i mask
      if (lane == 16 + i) him  = m32;   // lanes 16..23 keep row m0+8+i mask
    }
    if (lane < 8) {
      int row = m0 + lane;
      atomicOr(&hs[(((size_t)t * NEXP + e) * NTOK + row) * 16 + wordcol],
               (lowm & 0xFFFFu) << sh);
    } else if (lane >= 16 && lane < 24) {
      int row = m0 + 8 + (lane - 16);
      atomicOr(&hs[(((size_t)t * NEXP + e) * NTOK + row) * 16 + wordcol],
               (him >> 16) << sh);
    }
  }
}

// ---------------------------------------------------------------------------
// GEMM2 + LIF + combine: o[t] = hs[t,e] @ w_down[e]; LIF over t; weighted
// accumulate by routing weight; dense write of out. Experts with zero weight
// across the whole 32-token tile are skipped (identical forward value).
// Grid: (N/32, D/32). Block 128 = 4 waves of 16x16 tiles. K = ef = 512.
// ---------------------------------------------------------------------------
__global__ __launch_bounds__(128)
void k_down(const unsigned* __restrict__ hs, const float* __restrict__ wdown,
            const float* __restrict__ wroute, float* __restrict__ out) {
  __shared__ float Wd[32][33];  // [d][k] transposed
  __shared__ float wtok[32];
  __shared__ int skipflag;

  const int tid  = threadIdx.x;
  const int lane = tid & 31;
  const int wave = tid >> 5;
  const int wm = wave & 1, wn = wave >> 1;
  const int mb = blockIdx.x * 32;
  const int db = blockIdx.y * 32;
  const int m0 = mb + wm * 16;
  const int d0 = db + wn * 16;
  const int row16 = lane & 15;
  const int koff  = (lane >> 4) * 2;
  const int rowadd = (lane >= 16) ? 8 : 0;

  v8f acc[T_STEPS];
#pragma unroll
  for (int t = 0; t < T_STEPS; ++t)
#pragma unroll
    for (int i = 0; i < 8; ++i) acc[t][i] = 0.f;

  for (int e = 0; e < NEXP; ++e) {
    __syncthreads();  // protect wtok/Wd reads of previous iteration
    if (tid < 32) wtok[tid] = wroute[(size_t)(mb + tid) * NEXP + e];
    __syncthreads();
    if (tid == 0) {
      int f = 0;
      for (int j = 0; j < 32; ++j) f |= (wtok[j] != 0.f);
      skipflag = f;
    }
    __syncthreads();
    if (!skipflag) continue;  // uniform across the workgroup

    v8f o[T_STEPS];
#pragma unroll
    for (int t = 0; t < T_STEPS; ++t)
#pragma unroll
      for (int i = 0; i < 8; ++i) o[t][i] = 0.f;

    for (int kb = 0; kb < EFD; kb += 32) {
      __syncthreads();
      for (int idx = tid; idx < 1024; idx += 128) {  // 32 k * 32 d
        int k = idx >> 5, d = idx & 31;
        const float* p = &wdown[((size_t)e * EFD + kb + k) * DDIM + db + d];
        Wd[d][k] = *p;
        __builtin_prefetch(p + 32 * DDIM, 0, 1);     // next K-slab -> L2
      }
      __syncthreads();
      unsigned wbits[T_STEPS];
#pragma unroll
      for (int t = 0; t < T_STEPS; ++t)
        wbits[t] = hs[(((size_t)t * NEXP + e) * NTOK + (m0 + row16)) * 16 +
                      (kb >> 5)];
#pragma unroll
      for (int kk = 0; kk < 32; kk += 4) {
        v2f b;
        b.x = Wd[wn * 16 + row16][kk + koff];
        b.y = Wd[wn * 16 + row16][kk + koff + 1];
        const int kbit = kk + koff;
#pragma unroll
        for (int t = 0; t < T_STEPS; ++t) {
          v2f a;  // expand spike bits to {0.0f, 1.0f}
          a.x = (float)((wbits[t] >> kbit) & 1u);
          a.y = (float)((wbits[t] >> (kbit + 1)) & 1u);
          o[t] = wmma4(a, b, o[t]);
        }
      }
    }

    // LIF over t + routing-weighted accumulation
    float vmem[8];
#pragma unroll
    for (int i = 0; i < 8; ++i) vmem[i] = 0.f;
#pragma unroll
    for (int t = 0; t < T_STEPS; ++t) {
#pragma unroll
      for (int i = 0; i < 8; ++i) {
        float v = kBETA * vmem[i] + o[t][i];
        int s = (v >= kVTH);
        vmem[i] = v - (s ? kVTH : 0.f);
        if (s) acc[t][i] += wtok[wm * 16 + i + rowadd];
      }
    }
  }

  // dense store of out[t][n][d] (non-temporal: written once, never re-read)
  const int col = d0 + row16;
#pragma unroll
  for (int t = 0; t < T_STEPS; ++t)
#pragma unroll
    for (int i = 0; i < 8; ++i) {
      int row = m0 + i + rowadd;
      __builtin_nontemporal_store(acc[t][i],
                                  &out[((size_t)t * NTOK + row) * DDIM + col]);
    }
}

// ---------------------------------------------------------------------------
// Load-balance loss: lb = E * sum_e (topcnt[e]/(N*K)) * (rpsum[e]/N)
// ---------------------------------------------------------------------------
__global__ void k_lb(const float* __restrict__ stats, float* __restrict__ out) {
  if (threadIdx.x == 0 && blockIdx.x == 0) {
    float lb = 0.f;
    for (int e = 0; e < NEXP; ++e)
      lb += (stats[e] / (2.f * NTOK)) * (stats[8 + e] / (float)NTOK);
    out[(size_t)T_STEPS * NTOK * DDIM] = 8.f * lb;
  }
}

// ---------------------------------------------------------------------------
extern "C" void kernel_launch(void* const* d_in, const int* in_sizes, int n_in,
                              void* d_out, int out_size, void* d_ws,
                              size_t ws_size, hipStream_t stream) {
  (void)in_sizes; (void)n_in; (void)out_size; (void)ws_size;
  const float* x     = (const float*)d_in[0];  // (T,B,S,D) fp32
  const float* wup   = (const float*)d_in[1];  // (E,D,ef) fp32
  const float* wdown = (const float*)d_in[2];  // (E,ef,D) fp32
  const float* bias  = (const float*)d_in[3];  // (E,) fp32
  float* out = (float*)d_out;                  // (T,B,S,D) flat + lb scalar

  // Workspace layout: [hs bitmask 4MB][wroute N*E f32][stats 16 f32]
  unsigned* hs   = (unsigned*)d_ws;
  float* wroute  = (float*)((char*)d_ws + (size_t)HS_WORDS * 4);
  float* stats   = wroute + (size_t)NTOK * NEXP;

  k_init<<<1024, 256, 0, stream>>>(hs, stats);
  k_route<<<NTOK / 8, 256, 0, stream>>>(x, bias, wroute, stats);

  dim3 g1(NTOK / 32, EFD / 32, NEXP);
  k_up<<<g1, 128, 0, stream>>>(x, wup, hs);

  dim3 g2(NTOK / 32, DDIM / 32);
  k_down<<<g2, 128, 0, stream>>>(hs, wdown, wroute, out);

  k_lb<<<1, 32, 0, stream>>>(stats, out);
}